// GraphSAGELayer_22565758173856
// MI455X (gfx1250) — compile-verified
//
#include <hip/hip_runtime.h>

// GraphSAGE layer for MI455X (gfx1250, wave32).
// Phase 1: zero agg workspace. Phase 2: edge scatter-sum via L2-resident
// f32 atomics (feat=25.6MB, agg=12.8MB both fit the 192MB L2).
// Phase 3: fused dual GEMM out = feat[:ND]*w1^T + (agg/in_norm)*w2^T + b1 + b2
// using V_WMMA_F32_16X16X4_F32 (keeps full f32 precision; GEMM is only
// ~1.6 GFLOP so it is memory-bound and low precision buys nothing).

#define D 128  // D_IN == D_OUT == 128

typedef __attribute__((ext_vector_type(2))) float v2f;
typedef __attribute__((ext_vector_type(8))) float v8f;

// ---------------------------------------------------------------- zero agg
__global__ void sage_zero(float4* __restrict__ p, int n4) {
  int stride = gridDim.x * blockDim.x;
  float4 z = make_float4(0.f, 0.f, 0.f, 0.f);
  for (int i = blockIdx.x * blockDim.x + threadIdx.x; i < n4; i += stride)
    p[i] = z;
}

// ------------------------------------------------------------- scatter sum
// One wave32 per edge: lane L moves feat[src[e]][4L .. 4L+3] (full 512B row
// per edge, coalesced) and accumulates with global_atomic_add_f32 into agg.
__global__ void sage_scatter(const float* __restrict__ feat,
                             const int* __restrict__ src,
                             const int* __restrict__ dst,
                             float* __restrict__ agg, int E) {
  int tid  = blockIdx.x * blockDim.x + threadIdx.x;
  int e    = tid >> 5;
  int lane = tid & 31;
  if (e >= E) return;
  int s = src[e];
  int d = dst[e];
  const float4* frow = (const float4*)(feat + (size_t)s * D);
  float4 v = frow[lane];
  float* arow = agg + (size_t)d * D + lane * 4;
  atomicAdd(arow + 0, v.x);
  atomicAdd(arow + 1, v.y);
  atomicAdd(arow + 2, v.z);
  atomicAdd(arow + 3, v.w);
}

// ---------------------------------------------------------------- fused GEMM
// Block = 256 threads = 8 waves. Block b computes output rows [16b,16b+16),
// wave w computes output cols [16w,16w+16). K=128 -> 32 WMMA f32 16x16x4
// steps per GEMM, 64 total, one shared f32 accumulator.
//
// A frag (16x4 f32): lane l holds M=l&15, K = kbase + 2*(l>>4) + {0,1}  -> float2
// B frag (4x16 f32): lane l holds N=l&15, K = kbase + 2*(l>>4) + {0,1}  -> float2
//   (B column o is w[o, :], contiguous since w is [O,D] row-major)
// C/D (16x16 f32):   vgpr r -> M = 8*(l>>4)+r, N = l&15
__global__ void __launch_bounds__(256)
sage_gemm(const float* __restrict__ feat,
          const float* __restrict__ agg,
          const float* __restrict__ in_norm,
          const float* __restrict__ w1, const float* __restrict__ b1,
          const float* __restrict__ w2, const float* __restrict__ b2,
          float* __restrict__ out, int ND) {
  const int wave  = threadIdx.x >> 5;   // 0..7 : output-column tile
  const int lane  = threadIdx.x & 31;
  const int nlo   = lane & 15;
  const int khalf = lane >> 4;          // 0 -> K{0,1}, 1 -> K{2,3}
  const int row_base = blockIdx.x * 16;

  int arow = row_base + nlo;            // clamp so EXEC stays all-ones for WMMA
  if (arow >= ND) arow = ND - 1;
  const int o = wave * 16 + nlo;        // this lane's output column (B column)

  const float2* fA  = (const float2*)(feat + (size_t)arow * D);
  const float2* gA  = (const float2*)(agg  + (size_t)arow * D);
  const float2* w1B = (const float2*)(w1 + (size_t)o * D);
  const float2* w2B = (const float2*)(w2 + (size_t)o * D);
  const float inv = 1.0f / in_norm[arow];

  v8f c = {};

  // out += feat * w1^T
#pragma unroll
  for (int k = 0; k < D; k += 4) {
    float2 af = fA[(k >> 1) + khalf];
    float2 bf = w1B[(k >> 1) + khalf];
    v2f a, b;
    a[0] = af.x; a[1] = af.y;
    b[0] = bf.x; b[1] = bf.y;
    c = __builtin_amdgcn_wmma_f32_16x16x4_f32(false, a, false, b,
                                              (short)0, c, false, false);
  }
  // out += (agg / in_norm) * w2^T   (normalization folded into A fragment)
#pragma unroll
  for (int k = 0; k < D; k += 4) {
    float2 af = gA[(k >> 1) + khalf];
    float2 bf = w2B[(k >> 1) + khalf];
    v2f a, b;
    a[0] = af.x * inv; a[1] = af.y * inv;
    b[0] = bf.x;       b[1] = bf.y;
    c = __builtin_amdgcn_wmma_f32_16x16x4_f32(false, a, false, b,
                                              (short)0, c, false, false);
  }

  const float bias = b1[o] + b2[o];

#pragma unroll
  for (int r = 0; r < 8; ++r) {
    int m = khalf * 8 + r;
    int n = row_base + m;
    if (n < ND) out[(size_t)n * D + o] = c[r] + bias;  // predicate only store
  }
}

// ------------------------------------------------------------------ launch
extern "C" void kernel_launch(void* const* d_in, const int* in_sizes, int n_in,
                              void* d_out, int out_size, void* d_ws, size_t ws_size,
                              hipStream_t stream) {
  const float* feat    = (const float*)d_in[0];
  const float* in_norm = (const float*)d_in[1];
  const float* w1      = (const float*)d_in[2];
  const float* b1      = (const float*)d_in[3];
  const float* w2      = (const float*)d_in[4];
  const float* b2      = (const float*)d_in[5];
  const int*   src     = (const int*)d_in[6];
  const int*   dst     = (const int*)d_in[7];
  float*       out     = (float*)d_out;

  const int ND = in_sizes[1];          // num_dst  (in_norm is [num_dst])
  const int E  = in_sizes[6];          // number of edges

  float* agg = (float*)d_ws;           // ND * D floats of scratch

  // 1) zero the aggregation buffer (workspace is poisoned by the harness)
  {
    int n4 = (ND * D) / 4;
    int blocks = (n4 + 255) / 256;
    if (blocks > 4096) blocks = 4096;
    sage_zero<<<blocks, 256, 0, stream>>>((float4*)agg, n4);
  }
  // 2) edge scatter-sum: one wave per edge
  {
    long long threads = (long long)E * 32;
    int blocks = (int)((threads + 255) / 256);
    sage_scatter<<<blocks, 256, 0, stream>>>(feat, src, dst, agg, E);
  }
  // 3) fused dual GEMM + bias epilogue
  {
    int blocks = (ND + 15) / 16;
    sage_gemm<<<blocks, 256, 0, stream>>>(feat, agg, in_norm,
                                          w1, b1, w2, b2, out, ND);
  }
}